// pointconvbackbone_14654428414276
// MI455X (gfx1250) — compile-verified
//
#include <hip/hip_runtime.h>
#include <hip/hip_bf16.h>

typedef float v2f __attribute__((ext_vector_type(2)));
typedef float v8f __attribute__((ext_vector_type(8)));

// ---------------- transpose (B,3,N) -> (B,N,3) ----------------
__global__ void k_tr_b3n(const float* __restrict__ s, float* __restrict__ d, int N, int total) {
  int t = blockIdx.x * blockDim.x + threadIdx.x;
  if (t >= total) return;
  int b = t / (N * 3);
  int r = t - b * (N * 3);
  int n = r / 3, c = r - n * 3;
  d[t] = s[((size_t)b * 3 + c) * N + n];
}

// ---------------- density: inv_dens = 2.5*bw*N / sum_j exp(-d2/(2bw^2)) ----------------
__global__ void k_density(const float* __restrict__ xyz, float* __restrict__ invd, int N, float bw) {
  int b = blockIdx.y;
  int i = blockIdx.x * blockDim.x + threadIdx.x;
  const float* X = xyz + (size_t)b * N * 3;
  __shared__ float tx[128], ty[128], tz[128];
  float px = 0.f, py = 0.f, pz = 0.f;
  if (i < N) { px = X[i * 3]; py = X[i * 3 + 1]; pz = X[i * 3 + 2]; }
  const float cc = -1.0f / (2.0f * bw * bw);
  float acc = 0.f;
  for (int j0 = 0; j0 < N; j0 += 128) {
    int jj = j0 + (int)threadIdx.x;
    if (jj < N) { tx[threadIdx.x] = X[jj * 3]; ty[threadIdx.x] = X[jj * 3 + 1]; tz[threadIdx.x] = X[jj * 3 + 2]; }
    __syncthreads();
    int cnt = (N - j0 < 128) ? (N - j0) : 128;
    for (int j = 0; j < cnt; ++j) {
      float dx = px - tx[j], dy = py - ty[j], dz = pz - tz[j];
      acc += __expf(cc * (dx * dx + dy * dy + dz * dz));
    }
    __syncthreads();
  }
  if (i < N) invd[(size_t)b * N + i] = (2.5f * bw * (float)N) / acc;
}

// ---------------- farthest point sampling: one block per batch ----------------
__global__ void k_fps(const float* __restrict__ xyz, int* __restrict__ idx, int N, int npoint) {
  int b = blockIdx.x, tid = threadIdx.x;
  const float* X = xyz + (size_t)b * N * 3;
  __shared__ float dist[4096];
  __shared__ float rv[256];
  __shared__ int   ri[256];
  __shared__ int   s_far;
  for (int j = tid; j < N; j += 256) dist[j] = 1e10f;
  if (tid == 0) s_far = 0;
  __syncthreads();
  for (int i = 0; i < npoint; ++i) {
    int far = s_far;
    if (tid == 0) idx[(size_t)b * npoint + i] = far;
    float cx = X[far * 3], cy = X[far * 3 + 1], cz = X[far * 3 + 2];
    float best = -1.f; int bi = N;
    for (int j = tid; j < N; j += 256) {
      float dx = X[j * 3] - cx, dy = X[j * 3 + 1] - cy, dz = X[j * 3 + 2] - cz;
      float d = dx * dx + dy * dy + dz * dz;
      float nd = fminf(dist[j], d);
      dist[j] = nd;
      if (nd > best) { best = nd; bi = j; }
    }
    rv[tid] = best; ri[tid] = bi;
    __syncthreads();
    for (int s = 128; s > 0; s >>= 1) {
      if (tid < s) {
        float ov = rv[tid + s]; int oi = ri[tid + s];
        if (ov > rv[tid] || (ov == rv[tid] && oi < ri[tid])) { rv[tid] = ov; ri[tid] = oi; }
      }
      __syncthreads();
    }
    if (tid == 0) s_far = ri[0];
    __syncthreads();
  }
}

// ---------------- gather sampled coords ----------------
__global__ void k_gather_nx(const float* __restrict__ xyz, const int* __restrict__ fps,
                            float* __restrict__ nx, int N, int S, int total) {
  int t = blockIdx.x * blockDim.x + threadIdx.x;
  if (t >= total) return; // total = B*S*3
  int c = t % 3; int bs = t / 3; int b = bs / S; int s = bs - b * S;
  nx[t] = xyz[((size_t)b * N + fps[(size_t)b * S + s]) * 3 + c];
}

// ---------------- kNN (smallest nsample dists, tie -> lower index) ----------------
__global__ void k_knn(const float* __restrict__ q, const float* __restrict__ xyz,
                      int* __restrict__ knn, int S, int N, int nsample) {
  int bs = blockIdx.x; int b = bs / S;
  int tid = threadIdx.x;
  __shared__ float sd[4096];
  __shared__ float rv[128];
  __shared__ int   ri[128];
  __shared__ float s_pd; __shared__ int s_pi;
  const float* Q = q + (size_t)bs * 3;
  float qx = Q[0], qy = Q[1], qz = Q[2];
  const float* X = xyz + (size_t)b * N * 3;
  for (int j = tid; j < N; j += 128) {
    float dx = X[j * 3] - qx, dy = X[j * 3 + 1] - qy, dz = X[j * 3 + 2] - qz;
    sd[j] = dx * dx + dy * dy + dz * dz;
  }
  __syncthreads();
  float pd = -3.4e38f; int pix = -1;
  for (int t = 0; t < nsample; ++t) {
    float best = 3.4e38f; int bi = N;
    for (int j = tid; j < N; j += 128) {
      float dj = sd[j];
      bool gt = (dj > pd) || (dj == pd && j > pix);
      if (gt && (dj < best || (dj == best && j < bi))) { best = dj; bi = j; }
    }
    rv[tid] = best; ri[tid] = bi;
    __syncthreads();
    for (int s = 64; s > 0; s >>= 1) {
      if (tid < s) {
        float ov = rv[tid + s]; int oi = ri[tid + s];
        if (ov < rv[tid] || (ov == rv[tid] && oi < ri[tid])) { rv[tid] = ov; ri[tid] = oi; }
      }
      __syncthreads();
    }
    if (tid == 0) { knn[(size_t)bs * nsample + t] = ri[0]; s_pd = rv[0]; s_pi = ri[0]; }
    __syncthreads();
    pd = s_pd; pix = s_pi;
  }
}

// ---------------- grouping: gxyz (rel coords), gpts = [gxyz|points], gdens ----------------
__global__ void k_group(const float* __restrict__ xyz, const float* __restrict__ pts,
                        const float* __restrict__ invd, const int* __restrict__ knn,
                        const float* __restrict__ nx,
                        float* __restrict__ gxyz, float* __restrict__ gpts, float* __restrict__ gdens,
                        int S, int K, int N, int Cin, int total) {
  int t = blockIdx.x * blockDim.x + threadIdx.x;
  if (t >= total) return; // total = B*S*K
  int bs = t / K; int b = bs / S;
  int j = knn[(size_t)t];
  const float* Qc = nx + (size_t)bs * 3;
  const float* Pj = xyz + ((size_t)b * N + j) * 3;
  float gx = Pj[0] - Qc[0], gy = Pj[1] - Qc[1], gz = Pj[2] - Qc[2];
  gxyz[(size_t)t * 3] = gx; gxyz[(size_t)t * 3 + 1] = gy; gxyz[(size_t)t * 3 + 2] = gz;
  int Ct = 3 + Cin;
  float* gp = gpts + (size_t)t * Ct;
  gp[0] = gx; gp[1] = gy; gp[2] = gz;
  const float* pp = pts + ((size_t)b * N + j) * Cin;
  for (int c = 0; c < Cin; ++c) gp[3 + c] = pp[c];
  gdens[t] = invd[(size_t)b * N + j];
}

// ---------------- group_all (sa3): gxyz = xyz (absolute) ----------------
__global__ void k_group_all(const float* __restrict__ xyz, const float* __restrict__ pts,
                            const float* __restrict__ invd,
                            float* __restrict__ gxyz, float* __restrict__ gpts, float* __restrict__ gdens,
                            int K, int Cin, int total) {
  int t = blockIdx.x * blockDim.x + threadIdx.x;
  if (t >= total) return; // total = B*K (contiguous)
  const float* Pj = xyz + (size_t)t * 3;
  float gx = Pj[0], gy = Pj[1], gz = Pj[2];
  gxyz[(size_t)t * 3] = gx; gxyz[(size_t)t * 3 + 1] = gy; gxyz[(size_t)t * 3 + 2] = gz;
  int Ct = 3 + Cin;
  float* gp = gpts + (size_t)t * Ct;
  gp[0] = gx; gp[1] = gy; gp[2] = gz;
  const float* pp = pts + (size_t)t * Cin;
  for (int c = 0; c < Cin; ++c) gp[3 + c] = pp[c];
  gdens[t] = invd[t];
}

// ---------------- fused GEMM + bias + BN + ReLU via V_WMMA_F32_16X16X4_F32 ----------------
// C[M,N] = relu(((A@W)+bias)*g+beta). Block = 256 thr = 8 waves -> 128x32 macro-tile.
// W chunks (GKC x 32) are register-staged then committed to LDS; the next chunk's
// global loads are issued before the current chunk's WMMA compute (software pipeline).
#define GKC 64
__global__ __launch_bounds__(256) void k_wmma_gemm(
    const float* __restrict__ A, const float* __restrict__ W,
    const float* __restrict__ bias, const float* __restrict__ gam, const float* __restrict__ bet,
    float* __restrict__ C, int M, int K, int N) {
  __shared__ float sw[32 * (GKC + 4)];
  const int tid  = threadIdx.x;
  const int wave = tid >> 5, lane = tid & 31;
  const int l = lane & 15, hi = lane >> 4;
  const int tm = blockIdx.x * 8 + wave;
  const int col0 = blockIdx.y * 32;
  const int arow = tm * 16 + l;
  const bool activeM = (tm * 16) < M;
  const bool fullM   = (tm * 16 + 16) <= M;
  const bool fullN   = (col0 + 32) <= N;
  const float* Ar = A + (size_t)arow * K;   // deref only when row valid
  v8f acc0 = {0.f, 0.f, 0.f, 0.f, 0.f, 0.f, 0.f, 0.f};
  v8f acc1 = acc0;

  // each thread owns 8 elements of the 64x32 chunk
  const int kk0 = tid >> 5;        // 0..7  (row stride 8 below)
  const int cc0 = tid & 31;        // column
  float stage[8];

  auto fill_regs = [&](int k0) {
    if (fullN && (k0 + GKC <= K)) {
      #pragma unroll
      for (int i = 0; i < 8; ++i)
        stage[i] = W[(size_t)(k0 + kk0 + i * 8) * N + (col0 + cc0)];
    } else {
      #pragma unroll
      for (int i = 0; i < 8; ++i) {
        int gk = k0 + kk0 + i * 8, gc = col0 + cc0;
        stage[i] = (gk < K && gc < N) ? W[(size_t)gk * N + gc] : 0.f;
      }
    }
  };

  fill_regs(0);
  for (int k0 = 0; k0 < K; k0 += GKC) {
    // commit staged chunk to LDS (column-major, +4 pad)
    #pragma unroll
    for (int i = 0; i < 8; ++i)
      sw[cc0 * (GKC + 4) + kk0 + i * 8] = stage[i];
    __syncthreads();
    // issue next chunk's global loads; latency hidden under the WMMA compute below
    if (k0 + GKC < K) fill_regs(k0 + GKC);
    if (activeM) {
      const float* s0 = &sw[(size_t)l * (GKC + 4)];
      const float* s1 = &sw[(size_t)(l + 16) * (GKC + 4)];
      if (fullM && (k0 + GKC <= K)) {
        // fast path: straight-line loads, no guards
        #pragma unroll 4
        for (int kk = 0; kk < GKC; kk += 4) {
          int ka = kk + hi * 2;
          v2f a, b0, b1;
          a.x  = Ar[k0 + ka];
          a.y  = Ar[k0 + ka + 1];
          b0.x = s0[ka]; b0.y = s0[ka + 1];
          b1.x = s1[ka]; b1.y = s1[ka + 1];
          acc0 = __builtin_amdgcn_wmma_f32_16x16x4_f32(false, a, false, b0, (short)0, acc0, false, false);
          acc1 = __builtin_amdgcn_wmma_f32_16x16x4_f32(false, a, false, b1, (short)0, acc1, false, false);
        }
      } else {
        // guarded tail / partial-M path (B side already zero-padded in LDS)
        for (int kk = 0; kk < GKC; kk += 4) {
          int ka = kk + hi * 2;
          int gk = k0 + ka;
          v2f a = {0.f, 0.f}, b0, b1;
          bool rv = arow < M;
          if (rv && gk < K)     a.x = Ar[gk];
          if (rv && gk + 1 < K) a.y = Ar[gk + 1];
          b0.x = s0[ka]; b0.y = s0[ka + 1];
          b1.x = s1[ka]; b1.y = s1[ka + 1];
          acc0 = __builtin_amdgcn_wmma_f32_16x16x4_f32(false, a, false, b0, (short)0, acc0, false, false);
          acc1 = __builtin_amdgcn_wmma_f32_16x16x4_f32(false, a, false, b1, (short)0, acc1, false, false);
        }
      }
    }
    __syncthreads();
  }

  if (!activeM) return;
  for (int sub = 0; sub < 2; ++sub) {
    int bcol = col0 + sub * 16 + l;
    if (bcol >= N) continue;
    v8f acc = (sub == 0) ? acc0 : acc1;
    float bb = bias[bcol], gg = gam[bcol], be = bet[bcol];
    for (int r = 0; r < 8; ++r) {
      int row = tm * 16 + r + hi * 8;
      if (row < M) {
        float v = (acc[r] + bb) * gg + be;
        C[(size_t)row * N + bcol] = fmaxf(v, 0.f);
      }
    }
  }
}

// ---------------- h *= gdens / max_k(gdens), per (b,s) ----------------
__global__ void k_dens_scale(float* __restrict__ h, const float* __restrict__ gdens, int K, int C) {
  int bs = blockIdx.x, tid = threadIdx.x;
  __shared__ float red[256];
  const float* gd = gdens + (size_t)bs * K;
  float m = -3.4e38f;
  for (int k = tid; k < K; k += 256) m = fmaxf(m, gd[k]);
  red[tid] = m; __syncthreads();
  for (int s = 128; s > 0; s >>= 1) { if (tid < s) red[tid] = fmaxf(red[tid], red[tid + s]); __syncthreads(); }
  float inv = 1.0f / red[0];
  float* H = h + (size_t)bs * (size_t)K * C;
  for (int k = 0; k < K; ++k) {
    float sc = gd[k] * inv;
    float* Hr = H + (size_t)k * C;
    for (int c = tid; c < C; c += 256) Hr[c] *= sc;
  }
}

// ---------------- einsum bskc,bskw->bscw: per (b,s): (CxK)@(Kx16), out[c*16+w] ----------------
// Block = 256 thr = 8 waves; weightnet tile (Kx16, K<=128) register-staged into LDS once.
__global__ __launch_bounds__(256) void k_wmma_einsum(
    const float* __restrict__ h, const float* __restrict__ wn, float* __restrict__ out, int K, int C) {
  __shared__ float sw[16 * 132];            // column-major sw[w*(K+4)+k], K<=128
  int bs = blockIdx.x;
  int Kp = K + 4;
  const float* Wn = wn + (size_t)bs * (size_t)K * 16;
  {
    float tmp[8];
    int cnt = (K * 16) >> 8;                // K in {32,64,128} -> 2,4,8
    #pragma unroll 8
    for (int i = 0; i < cnt; ++i) tmp[i] = Wn[threadIdx.x + (i << 8)];
    #pragma unroll 8
    for (int i = 0; i < cnt; ++i) {
      int t = threadIdx.x + (i << 8);
      int kk = t >> 4, c = t & 15;
      sw[c * Kp + kk] = tmp[i];
    }
  }
  __syncthreads();
  int wave = threadIdx.x >> 5, lane = threadIdx.x & 31;
  int l = lane & 15, hi = lane >> 4;
  int tc = blockIdx.y * 8 + wave;           // C is a multiple of 128 in this net
  const float* H = h + (size_t)bs * (size_t)K * C;
  const float* sb = &sw[(size_t)l * Kp];
  int crow = tc * 16 + l;
  v8f acc = {0.f, 0.f, 0.f, 0.f, 0.f, 0.f, 0.f, 0.f};
  #pragma unroll 4
  for (int k0 = 0; k0 < K; k0 += 4) {       // K in {32,64,128}
    int ka = k0 + hi * 2;
    v2f a, b;
    a.x = H[(size_t)ka * C + crow];         // A[c][k] = h[k][c]
    a.y = H[(size_t)(ka + 1) * C + crow];
    b.x = sb[ka]; b.y = sb[ka + 1];
    acc = __builtin_amdgcn_wmma_f32_16x16x4_f32(false, a, false, b, (short)0, acc, false, false);
  }
  float* O = out + (size_t)bs * (size_t)C * 16;
  for (int r = 0; r < 8; ++r) {
    int c = tc * 16 + r + hi * 8;
    O[(size_t)c * 16 + l] = acc[r];
  }
}

extern "C" void kernel_launch(void* const* d_in, const int* in_sizes, int n_in,
                              void* d_out, int out_size, void* d_ws, size_t ws_size,
                              hipStream_t stream) {
  (void)in_sizes; (void)n_in; (void)out_size; (void)ws_size;
  const int B = 8, N1 = 4096;
  const float* xyz_in  = (const float*)d_in[0];
  const float* feat_in = (const float*)d_in[1];

  // Param leaves: jax pytree order (dict keys sorted) within 'params':
  // sa1,sa2,sa3 -> bn_linear.{beta,g}, linear.{W,b}, mlp[i].{W,b,beta,g}x3, weightnet[i].{W,b,beta,g}x3
  struct SaP {
    const float *bn_beta, *bn_g, *linW, *linb;
    const float *mW[3], *mb[3], *mbeta[3], *mg[3];
    const float *wW[3], *wb[3], *wbeta[3], *wg[3];
  };
  SaP sa[3];
  int pi = 2;
  for (int s = 0; s < 3; ++s) {
    sa[s].bn_beta = (const float*)d_in[pi++];
    sa[s].bn_g    = (const float*)d_in[pi++];
    sa[s].linW    = (const float*)d_in[pi++];
    sa[s].linb    = (const float*)d_in[pi++];
    for (int L = 0; L < 3; ++L) {
      sa[s].mW[L]    = (const float*)d_in[pi++];
      sa[s].mb[L]    = (const float*)d_in[pi++];
      sa[s].mbeta[L] = (const float*)d_in[pi++];
      sa[s].mg[L]    = (const float*)d_in[pi++];
    }
    for (int L = 0; L < 3; ++L) {
      sa[s].wW[L]    = (const float*)d_in[pi++];
      sa[s].wb[L]    = (const float*)d_in[pi++];
      sa[s].wbeta[L] = (const float*)d_in[pi++];
      sa[s].wg[L]    = (const float*)d_in[pi++];
    }
  }

  // -------- workspace bump allocation --------
  char* wp = (char*)d_ws;
  auto allocf = [&](size_t n) { float* p = (float*)wp; wp += ((n * 4 + 255) & ~(size_t)255); return p; };
  auto alloci = [&](size_t n) { int*   p = (int*)wp;   wp += ((n * 4 + 255) & ~(size_t)255); return p; };

  float* xyz_t  = allocf((size_t)B * N1 * 3);
  float* feat_t = allocf((size_t)B * N1 * 3);
  float* invd   = allocf((size_t)B * N1);
  int*   fpsb   = alloci((size_t)B * 512);
  float* nx1    = allocf((size_t)B * 512 * 3);
  float* nx2    = allocf((size_t)B * 128 * 3);
  int*   knnb   = alloci((size_t)B * 16384);
  float* gxyz   = allocf((size_t)B * 512 * 32 * 3);
  float* gpts   = allocf((size_t)B * 128 * 64 * 131);
  float* gdens  = allocf((size_t)B * 512 * 32);
  float* hA     = allocf((size_t)16777216);
  float* hB     = allocf((size_t)16777216);
  float* wA     = allocf((size_t)2097152);
  float* wB     = allocf((size_t)2097152);
  float* pts1   = allocf((size_t)B * 512 * 128);
  float* pts2   = allocf((size_t)B * 128 * 256);

  auto cdiv = [](int a, int b) { return (a + b - 1) / b; };
  auto gemm = [&](const float* A, const float* Wm, const float* bias, const float* g,
                  const float* be, float* Co, int M, int K, int Nc) {
    k_wmma_gemm<<<dim3(cdiv(M, 128), cdiv(Nc, 32)), 256, 0, stream>>>(A, Wm, bias, g, be, Co, M, K, Nc);
  };

  // transpose inputs (B,3,N)->(B,N,3)
  {
    int tot = B * N1 * 3;
    k_tr_b3n<<<cdiv(tot, 256), 256, 0, stream>>>(xyz_in,  xyz_t,  N1, tot);
    k_tr_b3n<<<cdiv(tot, 256), 256, 0, stream>>>(feat_in, feat_t, N1, tot);
  }

  // ================= SA1: npoint=512, nsample=32, bw=0.1, mlp 6->64->64->128 =================
  {
    const int N = N1, S = 512, K = 32, Cin = 3, Co = 128;
    const SaP& P = sa[0];
    k_density<<<dim3(cdiv(N, 128), B), 128, 0, stream>>>(xyz_t, invd, N, 0.1f);
    k_fps<<<B, 256, 0, stream>>>(xyz_t, fpsb, N, S);
    k_gather_nx<<<cdiv(B * S * 3, 256), 256, 0, stream>>>(xyz_t, fpsb, nx1, N, S, B * S * 3);
    k_knn<<<B * S, 128, 0, stream>>>(nx1, xyz_t, knnb, S, N, K);
    k_group<<<cdiv(B * S * K, 256), 256, 0, stream>>>(xyz_t, feat_t, invd, knnb, nx1,
                                                      gxyz, gpts, gdens, S, K, N, Cin, B * S * K);
    int M = B * S * K; // 131072
    gemm(gpts, P.mW[0], P.mb[0], P.mg[0], P.mbeta[0], hA, M, 3 + Cin, 64);
    gemm(hA,   P.mW[1], P.mb[1], P.mg[1], P.mbeta[1], hB, M, 64, 64);
    gemm(hB,   P.mW[2], P.mb[2], P.mg[2], P.mbeta[2], hA, M, 64, Co);
    k_dens_scale<<<B * S, 256, 0, stream>>>(hA, gdens, K, Co);
    gemm(gxyz, P.wW[0], P.wb[0], P.wg[0], P.wbeta[0], wA, M, 3, 8);
    gemm(wA,   P.wW[1], P.wb[1], P.wg[1], P.wbeta[1], wB, M, 8, 8);
    gemm(wB,   P.wW[2], P.wb[2], P.wg[2], P.wbeta[2], wA, M, 8, 16);
    k_wmma_einsum<<<dim3(B * S, Co / 128), 256, 0, stream>>>(hA, wA, hB, K, Co);
    gemm(hB, P.linW, P.linb, P.bn_g, P.bn_beta, pts1, B * S, 16 * Co, Co);
  }

  // ================= SA2: npoint=128, nsample=64, bw=0.2, mlp 131->128->128->256 =================
  {
    const int N = 512, S = 128, K = 64, Cin = 128, Co = 256;
    const SaP& P = sa[1];
    k_density<<<dim3(cdiv(N, 128), B), 128, 0, stream>>>(nx1, invd, N, 0.2f);
    k_fps<<<B, 256, 0, stream>>>(nx1, fpsb, N, S);
    k_gather_nx<<<cdiv(B * S * 3, 256), 256, 0, stream>>>(nx1, fpsb, nx2, N, S, B * S * 3);
    k_knn<<<B * S, 128, 0, stream>>>(nx2, nx1, knnb, S, N, K);
    k_group<<<cdiv(B * S * K, 256), 256, 0, stream>>>(nx1, pts1, invd, knnb, nx2,
                                                      gxyz, gpts, gdens, S, K, N, Cin, B * S * K);
    int M = B * S * K; // 65536
    gemm(gpts, P.mW[0], P.mb[0], P.mg[0], P.mbeta[0], hA, M, 3 + Cin, 128);
    gemm(hA,   P.mW[1], P.mb[1], P.mg[1], P.mbeta[1], hB, M, 128, 128);
    gemm(hB,   P.mW[2], P.mb[2], P.mg[2], P.mbeta[2], hA, M, 128, Co);
    k_dens_scale<<<B * S, 256, 0, stream>>>(hA, gdens, K, Co);
    gemm(gxyz, P.wW[0], P.wb[0], P.wg[0], P.wbeta[0], wA, M, 3, 8);
    gemm(wA,   P.wW[1], P.wb[1], P.wg[1], P.wbeta[1], wB, M, 8, 8);
    gemm(wB,   P.wW[2], P.wb[2], P.wg[2], P.wbeta[2], wA, M, 8, 16);
    k_wmma_einsum<<<dim3(B * S, Co / 128), 256, 0, stream>>>(hA, wA, hB, K, Co);
    gemm(hB, P.linW, P.linb, P.bn_g, P.bn_beta, pts2, B * S, 16 * Co, Co);
  }

  // ================= SA3: group_all, bw=0.4, mlp 259->256->512->1024 =================
  {
    const int K = 128, Cin = 256, Co = 1024;
    const SaP& P = sa[2];
    k_density<<<dim3(1, B), 128, 0, stream>>>(nx2, invd, K, 0.4f);
    k_group_all<<<cdiv(B * K, 256), 256, 0, stream>>>(nx2, pts2, invd, gxyz, gpts, gdens, K, Cin, B * K);
    int M = B * K; // 1024
    gemm(gpts, P.mW[0], P.mb[0], P.mg[0], P.mbeta[0], hA, M, 3 + Cin, 256);
    gemm(hA,   P.mW[1], P.mb[1], P.mg[1], P.mbeta[1], hB, M, 256, 512);
    gemm(hB,   P.mW[2], P.mb[2], P.mg[2], P.mbeta[2], hA, M, 512, Co);
    k_dens_scale<<<B, 256, 0, stream>>>(hA, gdens, K, Co);
    gemm(gxyz, P.wW[0], P.wb[0], P.wg[0], P.wbeta[0], wA, M, 3, 8);
    gemm(wA,   P.wW[1], P.wb[1], P.wg[1], P.wbeta[1], wB, M, 8, 8);
    gemm(wB,   P.wW[2], P.wb[2], P.wg[2], P.wbeta[2], wA, M, 8, 16);
    k_wmma_einsum<<<dim3(B, Co / 128), 256, 0, stream>>>(hA, wA, hB, K, Co);
    gemm(hB, P.linW, P.linb, P.bn_g, P.bn_beta, (float*)d_out, B, 16 * Co, Co); // -> (8,1024)
  }
}